// StructuralDescriptor_62826781605927
// MI455X (gfx1250) — compile-verified
//
#include <hip/hip_runtime.h>
#include <hip/hip_bf16.h>
#include <cstddef>

// ---------------------------------------------------------------------------
// MeshNet structural descriptor for MI455X (gfx1250), wave32 + WMMA f32.
// All GEMMs run on v_wmma_f32_16x16x4_f32 (exact f32 MACs).
// ---------------------------------------------------------------------------

typedef __attribute__((ext_vector_type(2))) float v2f;
typedef __attribute__((ext_vector_type(8))) float v8f;

#define FDIM 8192
#define NTOT 65536   // B*F = 8*8192

enum { MODE_RAW = 0, MODE_BN = 1, MODE_CORNERS = 2 };

// --------------------------- B-operand loader ------------------------------
template <int MODE>
__device__ __forceinline__ float load_x(const float* __restrict__ X,
                                        const float* __restrict__ sc,
                                        const float* __restrict__ sh,
                                        int K, int c, int n, int slice) {
  if (c >= K) return 0.f;
  if (MODE == MODE_CORNERS) {
    // corners is [B, 9, F]; slice s uses channels (c + 3*s) mod 9
    int b = n >> 13, f = n & (FDIM - 1);
    int ch = c + 3 * slice;
    if (ch >= 9) ch -= 9;
    return X[((size_t)b * 9 + ch) * FDIM + f];
  } else {
    float v = X[(size_t)c * NTOT + n];
    if (MODE == MODE_BN) v = fmaxf(fmaf(v, sc[c], sh[c]), 0.f);
    return v;
  }
}

// ------------------- fused GEMM + bias + BN-stats kernel -------------------
// Y[M,N] = W[M,K] * Xn[K,N] + bias ; also accumulates per-row sum / sumsq.
// One wave = one 16x16 tile, K fully unrolled over 16x16x4 f32 WMMAs.
template <int MODE, int K, int M>
__global__ __launch_bounds__(256) void gemm_bn_kernel(
    const float* __restrict__ W, const float* __restrict__ bias,
    const float* __restrict__ X,
    const float* __restrict__ sc, const float* __restrict__ sh,
    float* __restrict__ Y, float* __restrict__ stSum, float* __restrict__ stSq,
    int slice) {
  __shared__ float Wlds[16 * K];  // the 16 W-rows this block needs
  const int mt = blockIdx.y;
  for (int i = threadIdx.x; i < 16 * K; i += 256) {
    int r = i / K, c = i - r * K;
    int row = mt * 16 + r;
    Wlds[i] = (row < M) ? W[(size_t)row * K + c] : 0.f;
  }
  __syncthreads();

  const int wave = threadIdx.x >> 5;
  const int lane = threadIdx.x & 31;
  const int half = lane >> 4;    // 0: lanes 0-15, 1: lanes 16-31
  const int l15  = lane & 15;
  const int n0   = (blockIdx.x * 8 + wave) * 16;
  const int ncol = n0 + l15;

  v8f acc = {};
#pragma unroll
  for (int k0 = 0; k0 < K; k0 += 4) {
    // A (16x4 f32): lanes 0-15 hold K=k0,k0+1 ; lanes 16-31 hold K=k0+2,k0+3
    int ka = k0 + half * 2;
    v2f a, b;
    a.x = (ka     < K) ? Wlds[l15 * K + ka]     : 0.f;
    a.y = (ka + 1 < K) ? Wlds[l15 * K + ka + 1] : 0.f;
    // B (4x16 f32): same K split across halves, N striped across lanes
    b.x = load_x<MODE>(X, sc, sh, K, ka,     ncol, slice);
    b.y = load_x<MODE>(X, sc, sh, K, ka + 1, ncol, slice);
    acc = __builtin_amdgcn_wmma_f32_16x16x4_f32(
        /*neg_a=*/false, a, /*neg_b=*/false, b,
        /*c_mod=*/(short)0, acc, /*reuse_a=*/false, /*reuse_b=*/false);
  }

  // Epilogue: bias, store, and per-row sum/sumsq (16-lane reduce + atomics).
#pragma unroll
  for (int e = 0; e < 8; ++e) {
    int row = mt * 16 + e + half * 8;  // C/D layout: VGPR e -> M = e (+8 hi half)
    if (row < M) {                     // uniform within each 16-lane group
      float v = acc[e] + bias[row];
      Y[(size_t)row * NTOT + ncol] = v;
      float s = v, q = v * v;
#pragma unroll
      for (int off = 1; off < 16; off <<= 1) {
        s += __shfl_xor(s, off, 32);
        q += __shfl_xor(q, off, 32);
      }
      if (l15 == 0) {
        atomicAdd(&stSum[row], s);
        atomicAdd(&stSq[row], q);
      }
    }
  }
}

// --------------------------- small helpers ---------------------------------
__global__ void zero_stats(float* __restrict__ p, int n) {
  int i = blockIdx.x * 256 + threadIdx.x;
  if (i < n) p[i] = 0.f;
}

// BN finalize: scale = g * rsqrt(var+eps), shift = beta - mean*scale
__global__ void finalize_bn(const float* __restrict__ sum,
                            const float* __restrict__ sq,
                            const float* __restrict__ g,
                            const float* __restrict__ be,
                            float* __restrict__ scale,
                            float* __restrict__ shift, int C) {
  int c = blockIdx.x * blockDim.x + threadIdx.x;
  if (c < C) {
    float mean = sum[c] * (1.f / (float)NTOT);
    float var  = sq[c]  * (1.f / (float)NTOT) - mean * mean;
    float s = g[c] * rsqrtf(var + 1e-5f);
    scale[c] = s;
    shift[c] = fmaf(-mean, s, be[c]);
  }
}

// facc (+)= relu(bn(x)) / 3   over [32, NTOT]
__global__ __launch_bounds__(256) void accum_norm(const float* __restrict__ X,
                                                  const float* __restrict__ sc,
                                                  const float* __restrict__ sh,
                                                  float* __restrict__ dst,
                                                  int first) {
  size_t i = (size_t)blockIdx.x * 256 + threadIdx.x;
  int c = (int)(i >> 16);
  float v = fmaxf(fmaf(X[i], sc[c], sh[c]), 0.f) * (1.f / 3.f);
  dst[i] = first ? v : (dst[i] + v);
}

// dst[c,n] = relu(bn(x[c,n]))  over [C, NTOT]
__global__ __launch_bounds__(256) void norm_relu(const float* __restrict__ X,
                                                 const float* __restrict__ sc,
                                                 const float* __restrict__ sh,
                                                 float* __restrict__ dst, int C) {
  size_t i = (size_t)blockIdx.x * 256 + threadIdx.x;
  if (i < (size_t)C * NTOT) {
    int c = (int)(i >> 16);
    dst[i] = fmaxf(fmaf(X[i], sc[c], sh[c]), 0.f);
  }
}

// concat rows 128..130 <- normals [B,3,F] remapped to [3, NTOT]
__global__ __launch_bounds__(256) void copy_normals(
    const float* __restrict__ normals, float* __restrict__ dst) {
  int i = blockIdx.x * 256 + threadIdx.x;  // 3*NTOT
  int c = i >> 16;
  int n = i & (NTOT - 1);
  int b = n >> 13, f = n & (FDIM - 1);
  dst[(size_t)c * NTOT + n] = normals[((size_t)b * 3 + c) * FDIM + f];
}

// final: relu(bn(x)) from [131, NTOT] into out [B, 131, F]
__global__ __launch_bounds__(256) void write_out(const float* __restrict__ X,
                                                 const float* __restrict__ sc,
                                                 const float* __restrict__ sh,
                                                 float* __restrict__ out) {
  size_t i = (size_t)blockIdx.x * 256 + threadIdx.x;
  if (i < (size_t)131 * NTOT) {
    int c = (int)(i >> 16);
    int n = (int)(i & (NTOT - 1));
    int b = n >> 13, f = n & (FDIM - 1);
    float v = fmaxf(fmaf(X[i], sc[c], sh[c]), 0.f);
    out[((size_t)b * 131 + c) * FDIM + f] = v;
  }
}

// sphere kernel directions: wdir[3][256], w2[256]  (256 = K*4)
__global__ void kc_weights(const float* __restrict__ alpha,
                           const float* __restrict__ beta,
                           float* __restrict__ wdir, float* __restrict__ w2) {
  int i = threadIdx.x;  // one block of 256
  float a = alpha[i], b = beta[i];
  float wx = sinf(a) * cosf(b);
  float wy = sinf(a) * sinf(b);
  float wz = cosf(a);
  wdir[i] = wx;
  wdir[256 + i] = wy;
  wdir[512 + i] = wz;
  w2[i] = wx * wx + wy * wy + wz * wz;
}

// Gaussian kernel correlation: feat[64, NTOT] + fused BN stats.
__global__ __launch_bounds__(256) void kernel_corr(
    const float* __restrict__ normals, const int* __restrict__ nidx,
    const float* __restrict__ wdir, const float* __restrict__ w2,
    float* __restrict__ feat, float* __restrict__ stSum,
    float* __restrict__ stSq) {
  int n = blockIdx.x * 256 + threadIdx.x;  // 65536 threads
  int b = n >> 13, f = n & (FDIM - 1);
  const float* nb = normals + (size_t)b * 3 * FDIM;
  float px[4], py[4], pz[4], p2[4];
  px[0] = nb[f];
  py[0] = nb[FDIM + f];
  pz[0] = nb[2 * FDIM + f];
#pragma unroll
  for (int j = 0; j < 3; ++j) {
    int idx = nidx[((size_t)b * FDIM + f) * 3 + j];
    px[j + 1] = nb[idx];
    py[j + 1] = nb[FDIM + idx];
    pz[j + 1] = nb[2 * FDIM + idx];
  }
#pragma unroll
  for (int p = 0; p < 4; ++p)
    p2[p] = px[p] * px[p] + py[p] * py[p] + pz[p] * pz[p];

  int lane = threadIdx.x & 31;
  for (int k = 0; k < 64; ++k) {
    float acc = 0.f;
#pragma unroll
    for (int q = 0; q < 4; ++q) {
      int w = k * 4 + q;
      float wx = wdir[w], wy = wdir[256 + w], wz = wdir[512 + w];
      float ww = w2[w];
#pragma unroll
      for (int p = 0; p < 4; ++p) {
        float d = p2[p] + ww - 2.f * (px[p] * wx + py[p] * wy + pz[p] * wz);
        acc += __expf(d * -12.5f);  // -1/(2*sigma^2), sigma = 0.2
      }
    }
    float v = acc * (1.f / 16.f);  // / (P*4)
    feat[(size_t)k * NTOT + n] = v;
    float s = v, qq = v * v;
#pragma unroll
    for (int off = 1; off < 32; off <<= 1) {
      s += __shfl_xor(s, off, 32);
      qq += __shfl_xor(qq, off, 32);
    }
    if (lane == 0) {
      atomicAdd(&stSum[k], s);
      atomicAdd(&stSq[k], qq);
    }
  }
}

// ------------------------------ host driver --------------------------------
// stats region (floats): per layer block = [sum C | sq C | scale C | shift C]
static constexpr size_t STAT_R1 = 0;            // 3 slices * 128
static constexpr size_t STAT_R2 = 384;          // 3 slices * 128
static constexpr size_t STAT_F1 = 768;          // 256
static constexpr size_t STAT_F2 = 1024;         // 256
static constexpr size_t STAT_KC = 1280;         // 256
static constexpr size_t STAT_S1 = 1536;         // 524
static constexpr size_t STAT_S2 = 2060;         // 524
static constexpr size_t STAT_END = 2584;
static constexpr size_t OFF_WDIR = 2592;        // 768 floats
static constexpr size_t OFF_W2   = OFF_WDIR + 768;
static constexpr size_t OFF_BIG  = 4096;

extern "C" void kernel_launch(void* const* d_in, const int* in_sizes, int n_in,
                              void* d_out, int out_size, void* d_ws,
                              size_t ws_size, hipStream_t stream) {
  // setup_inputs() insertion order
  const float* corners = (const float*)d_in[0];
  const float* normals = (const float*)d_in[1];
  const int*   nidx    = (const int*)d_in[2];
  const float* w_alpha = (const float*)d_in[3];
  const float* w_beta  = (const float*)d_in[4];
  const float* k_g  = (const float*)d_in[5];
  const float* k_be = (const float*)d_in[6];
  const float* r_w1 = (const float*)d_in[7];
  const float* r_b1 = (const float*)d_in[8];
  const float* r_g1 = (const float*)d_in[9];
  const float* r_be1 = (const float*)d_in[10];
  const float* r_w2 = (const float*)d_in[11];
  const float* r_b2 = (const float*)d_in[12];
  const float* r_g2 = (const float*)d_in[13];
  const float* r_be2 = (const float*)d_in[14];
  const float* f_w1 = (const float*)d_in[15];
  const float* f_b1 = (const float*)d_in[16];
  const float* f_g1 = (const float*)d_in[17];
  const float* f_be1 = (const float*)d_in[18];
  const float* f_w2 = (const float*)d_in[19];
  const float* f_b2 = (const float*)d_in[20];
  const float* f_g2 = (const float*)d_in[21];
  const float* f_be2 = (const float*)d_in[22];
  const float* s_w1 = (const float*)d_in[23];
  const float* s_b1 = (const float*)d_in[24];
  const float* s_g1 = (const float*)d_in[25];
  const float* s_be1 = (const float*)d_in[26];
  const float* s_w2 = (const float*)d_in[27];
  const float* s_b2 = (const float*)d_in[28];
  const float* s_g2 = (const float*)d_in[29];
  const float* s_be2 = (const float*)d_in[30];

  float* ws   = (float*)d_ws;
  float* wdir = ws + OFF_WDIR;
  float* w2   = ws + OFF_W2;
  float* bufA = ws + OFF_BIG;                    // 131*NTOT  (t1 / tf1 / t5)
  float* bufB = bufA + (size_t)131 * NTOT;       // 131*NTOT  (t2 / tf2 / t6)
  float* bufC = bufB + (size_t)131 * NTOT;       // 131*NTOT  (concat)
  float* bufD = bufC + (size_t)131 * NTOT;       // 64*NTOT   (feat)
  float* bufE = bufD + (size_t)64 * NTOT;        // 32*NTOT   (facc)

  const dim3 blk(256);

  // reset BN accumulators every call (graph-replay safe)
  zero_stats<<<dim3((STAT_END + 255) / 256), blk, 0, stream>>>(ws, (int)STAT_END);

  // kernel-correlation branch (independent of rotate branch)
  kc_weights<<<dim3(1), blk, 0, stream>>>(w_alpha, w_beta, wdir, w2);
  {
    float* st = ws + STAT_KC;
    kernel_corr<<<dim3(NTOT / 256), blk, 0, stream>>>(normals, nidx, wdir, w2,
                                                      bufD, st, st + 64);
  }

  // rotate MLP over the three corner slices, averaged into bufE
  for (int s = 0; s < 3; ++s) {
    float* st1 = ws + STAT_R1 + (size_t)s * 128;
    float* st2 = ws + STAT_R2 + (size_t)s * 128;
    gemm_bn_kernel<MODE_CORNERS, 6, 32><<<dim3(512, 2), blk, 0, stream>>>(
        r_w1, r_b1, corners, nullptr, nullptr, bufA, st1, st1 + 32, s);
    finalize_bn<<<1, 256, 0, stream>>>(st1, st1 + 32, r_g1, r_be1, st1 + 64,
                                       st1 + 96, 32);
    gemm_bn_kernel<MODE_BN, 32, 32><<<dim3(512, 2), blk, 0, stream>>>(
        r_w2, r_b2, bufA, st1 + 64, st1 + 96, bufB, st2, st2 + 32, 0);
    finalize_bn<<<1, 256, 0, stream>>>(st2, st2 + 32, r_g2, r_be2, st2 + 64,
                                       st2 + 96, 32);
    accum_norm<<<dim3((32 * NTOT) / 256), blk, 0, stream>>>(
        bufB, st2 + 64, st2 + 96, bufE, s == 0 ? 1 : 0);
  }

  // face conv 32->64->64
  {
    float* st = ws + STAT_F1;
    gemm_bn_kernel<MODE_RAW, 32, 64><<<dim3(512, 4), blk, 0, stream>>>(
        f_w1, f_b1, bufE, nullptr, nullptr, bufA, st, st + 64, 0);
    finalize_bn<<<1, 256, 0, stream>>>(st, st + 64, f_g1, f_be1, st + 128,
                                       st + 192, 64);
  }
  {
    float* st1 = ws + STAT_F1;
    float* st = ws + STAT_F2;
    gemm_bn_kernel<MODE_BN, 64, 64><<<dim3(512, 4), blk, 0, stream>>>(
        f_w2, f_b2, bufA, st1 + 128, st1 + 192, bufB, st, st + 64, 0);
    finalize_bn<<<1, 256, 0, stream>>>(st, st + 64, f_g2, f_be2, st + 128,
                                       st + 192, 64);
  }
  {
    float* st = ws + STAT_KC;
    finalize_bn<<<1, 256, 0, stream>>>(st, st + 64, k_g, k_be, st + 128,
                                       st + 192, 64);
  }

  // concat [sf1 | sf2 | normals] -> bufC (131 x NTOT)
  {
    float* stF2 = ws + STAT_F2;
    float* stKC = ws + STAT_KC;
    norm_relu<<<dim3((64 * NTOT) / 256), blk, 0, stream>>>(
        bufB, stF2 + 128, stF2 + 192, bufC, 64);
    norm_relu<<<dim3((64 * NTOT) / 256), blk, 0, stream>>>(
        bufD, stKC + 128, stKC + 192, bufC + (size_t)64 * NTOT, 64);
    copy_normals<<<dim3((3 * NTOT) / 256), blk, 0, stream>>>(
        normals, bufC + (size_t)128 * NTOT);
  }

  // structural convs 131->131->131
  {
    float* st = ws + STAT_S1;
    gemm_bn_kernel<MODE_RAW, 131, 131><<<dim3(512, 9), blk, 0, stream>>>(
        s_w1, s_b1, bufC, nullptr, nullptr, bufA, st, st + 131, 0);
    finalize_bn<<<1, 256, 0, stream>>>(st, st + 131, s_g1, s_be1, st + 262,
                                       st + 393, 131);
  }
  {
    float* st1 = ws + STAT_S1;
    float* st = ws + STAT_S2;
    gemm_bn_kernel<MODE_BN, 131, 131><<<dim3(512, 9), blk, 0, stream>>>(
        s_w2, s_b2, bufA, st1 + 262, st1 + 393, bufB, st, st + 131, 0);
    finalize_bn<<<1, 256, 0, stream>>>(st, st + 131, s_g2, s_be2, st + 262,
                                       st + 393, 131);
    write_out<<<dim3((131 * NTOT + 255) / 256), blk, 0, stream>>>(
        bufB, st + 262, st + 393, (float*)d_out);
  }
}